// graph_t_6983616824414
// MI455X (gfx1250) — compile-verified
//
#include <hip/hip_runtime.h>

// Reference:
//   pa = coords[edge_a]; pb = coords[edge_b]; v = pb - pa
//   len = |v|; dir = v/len; force_on_a = -f * dir
//   weight = (0, 0, -len * density * 0.5)
//   out = prev; out[edge_a] += force_on_a + weight; out[edge_b] += -force_on_a + weight
//
// Memory-bound irregular scatter-add: no matrix contraction -> WMMA inapplicable.
// gfx1250 paths used: wave32 blocks, returnless global_atomic_add_f32 (L2-side
// RMW, fire-and-forget), global_prefetch_b8 on streaming arrays, and NT
// temporal hints (load TH policy) so the 64MB of once-read edge data does not
// evict the 2.4MB atomic-hot coords/out working set from the 192MB L2.

// Native clang vector type: required by __builtin_nontemporal_load (HIP's
// float4 is a struct and is rejected by the builtin).
typedef float v4f __attribute__((ext_vector_type(4)));

__global__ void copy_prev_f4(const v4f* __restrict__ prev,
                             v4f* __restrict__ out, int n4) {
    int i = blockIdx.x * blockDim.x + threadIdx.x;
    if (i < n4) {
        // prev is read exactly once -> NT load; out stays RT (atomics hit it next).
        v4f v = __builtin_nontemporal_load(&prev[i]);
        out[i] = v;
    }
}

__global__ void edge_scatter(const float* __restrict__ coords,
                             const float* __restrict__ forces,
                             const float* __restrict__ density,
                             const long long* __restrict__ edge_a,
                             const long long* __restrict__ edge_b,
                             float* __restrict__ out,
                             int n_edges) {
    int e = blockIdx.x * blockDim.x + threadIdx.x;
    if (e >= n_edges) return;

    // Prefetch the streaming arrays ~32KB ahead (global_prefetch_b8).
    int pf = e + 4096;
    if (pf < n_edges) {
        __builtin_prefetch(&edge_a[pf], 0, 0);
        __builtin_prefetch(&edge_b[pf], 0, 0);
        __builtin_prefetch(&forces[pf], 0, 0);
    }

    // Streaming, read-once data: NT temporal hint keeps it from occupying L2.
    long long ia64 = __builtin_nontemporal_load(&edge_a[e]);
    long long ib64 = __builtin_nontemporal_load(&edge_b[e]);
    float f        = __builtin_nontemporal_load(&forces[e]);
    int ia = (int)ia64;                // values < N_POINTS, fit in i32
    int ib = (int)ib64;

    float dens = density[0];           // uniform -> scalarized s_load

    // Gathers: coords is 1.2MB, fully L2-resident (regular-temporal loads).
    float ax = coords[3 * ia + 0];
    float ay = coords[3 * ia + 1];
    float az = coords[3 * ia + 2];
    float bx = coords[3 * ib + 0];
    float by = coords[3 * ib + 1];
    float bz = coords[3 * ib + 2];

    float vx = bx - ax, vy = by - ay, vz = bz - az;
    float d2  = vx * vx + vy * vy + vz * vz;
    float len = sqrtf(d2);
    float s   = -f / len;              // -(f) * (1/len): fold negate into scale

    float fxa = s * vx;                // force_on_a
    float fya = s * vy;
    float fza = s * vz;
    float wz  = -0.5f * dens * len;    // weight z-component (x,y are zero)

    // 6 returnless f32 atomics per edge; z folds force + weight into one atomic.
    // No packed f32 atomic exists (PK_ADD is f16/bf16 only), so 6 is minimal
    // at full precision for arbitrary endpoint indices.
    atomicAdd(&out[3 * ia + 0],  fxa);
    atomicAdd(&out[3 * ia + 1],  fya);
    atomicAdd(&out[3 * ia + 2],  fza + wz);
    atomicAdd(&out[3 * ib + 0], -fxa);
    atomicAdd(&out[3 * ib + 1], -fya);
    atomicAdd(&out[3 * ib + 2], -fza + wz);
}

extern "C" void kernel_launch(void* const* d_in, const int* in_sizes, int n_in,
                              void* d_out, int out_size, void* d_ws, size_t ws_size,
                              hipStream_t stream) {
    // setup_inputs() order:
    //   0: coords  [N_POINTS*3] f32
    //   1: forces  [N_EDGES]    f32
    //   2: prev    [N_POINTS*3] f32
    //   3: density [1]          f32
    //   4: edge_a  [N_EDGES]    i64
    //   5: edge_b  [N_EDGES]    i64
    const float*     coords  = (const float*)d_in[0];
    const float*     forces  = (const float*)d_in[1];
    const float*     prev    = (const float*)d_in[2];
    const float*     density = (const float*)d_in[3];
    const long long* edge_a  = (const long long*)d_in[4];
    const long long* edge_b  = (const long long*)d_in[5];
    float*           out     = (float*)d_out;

    const int n_edges = in_sizes[1];
    const int n_flt   = in_sizes[2];   // N_POINTS*3 = 300000, divisible by 4

    // 1) out = prev (float4-vectorized init; d_out is poisoned by harness)
    {
        int n4 = n_flt / 4;            // 75000
        int threads = 256;
        int blocks = (n4 + threads - 1) / threads;
        copy_prev_f4<<<blocks, threads, 0, stream>>>(
            (const v4f*)prev, (v4f*)out, n4);
    }

    // 2) scatter-add all edge contributions
    {
        int threads = 256;             // 8 wave32s per block
        int blocks = (n_edges + threads - 1) / threads;
        edge_scatter<<<blocks, threads, 0, stream>>>(
            coords, forces, density, edge_a, edge_b, out, n_edges);
    }
}